// Attention_47871705481539
// MI455X (gfx1250) — compile-verified
//
#include <hip/hip_runtime.h>
#include <math.h>

// ---------------------------------------------------------------------------
// CDNA5 (gfx1250) fp32 attention pipeline built on V_WMMA_F32_16X16X4_F32.
// wave32 everywhere. GEMMs use 2x2 register blocking (32x32 per wave) to
// double WMMA/load density; attention stages K/V tiles through LDS, using
// the gfx1250 async global->LDS path (ASYNCcnt-tracked) plus prefetch.
// ---------------------------------------------------------------------------

typedef __attribute__((ext_vector_type(2))) float v2f;
typedef __attribute__((ext_vector_type(8))) float v8f;

#define HEADS    8
#define DHEAD    64
#define C_IN     256
#define PIX      4096          // 64*64 tokens
#define HIDDEN   512
#define QKV_ROWS 1536
#define LKV      4112          // 4096 tokens + 4 mem + 12 pad (257 tiles of 16)
#define KTILES   257

__device__ __forceinline__ v8f wmma_f32_k4(v2f a, v2f b, v8f c) {
  // (neg_a, A, neg_b, B, c_mod, C, reuse_a, reuse_b)
  return __builtin_amdgcn_wmma_f32_16x16x4_f32(false, a, false, b, (short)0, c,
                                               false, false);
}

#if __has_builtin(__builtin_amdgcn_global_load_async_to_lds_b32) && \
    __has_builtin(__builtin_amdgcn_s_wait_asynccnt)
#define HAVE_ASYNC_LDS 1
__device__ __forceinline__ void lds_async_f32(float* dst_lds, const float* src_g) {
  __builtin_amdgcn_global_load_async_to_lds_b32(
      (__attribute__((address_space(1))) int*)(unsigned long long)src_g,
      (__attribute__((address_space(3))) int*)(unsigned)(unsigned long long)dst_lds,
      0, 0);
}
#else
#define HAVE_ASYNC_LDS 0
#endif

// ---------------------------------------------------------------------------
// 1) RMSNorm over channels, per pixel.  x:[256][4096] -> xn:[256][4096]
// ---------------------------------------------------------------------------
__global__ void k_rmsnorm(const float* __restrict__ x,
                          const float* __restrict__ gamma,
                          float* __restrict__ xn) {
  const int p = blockIdx.x * blockDim.x + threadIdx.x;   // 4096 pixels exactly
  float ss = 0.f;
  for (int c = 0; c < C_IN; ++c) { float v = x[c * PIX + p]; ss += v * v; }
  const float inv = 16.0f * rsqrtf(ss);                  // sqrt(256)/||x||
  for (int c = 0; c < C_IN; ++c)
    xn[c * PIX + p] = x[c * PIX + p] * inv * (gamma[c] + 1.0f);
}

// ---------------------------------------------------------------------------
// 2) Memory-KV tokens into rows/cols 4096..4099, zero the pad.
//    Kt:[8][64][4112] (d-major, key contiguous)  V:[8][4112][64]
// ---------------------------------------------------------------------------
__global__ void k_initkv(const float* __restrict__ mem_kv,
                         float* __restrict__ Kt, float* __restrict__ V) {
  const int h = blockIdx.x;
  for (int idx = threadIdx.x; idx < 16 * DHEAD; idx += blockDim.x) {
    const int j = idx >> 6, d = idx & 63;                // key 4096+j, dim d
    const float kv = (j < 4) ? mem_kv[((0 * HEADS + h) * 4 + j) * DHEAD + d] : 0.f;
    const float vv = (j < 4) ? mem_kv[((1 * HEADS + h) * 4 + j) * DHEAD + d] : 0.f;
    Kt[(h * DHEAD + d) * LKV + 4096 + j] = kv;
    V [(h * LKV + 4096 + j) * DHEAD + d] = vv;
  }
}

// ---------------------------------------------------------------------------
// 3) QKV projection: qkv[1536][4096] = w_qkv[1536][256] @ xn[256][4096]
//    32x32 block per wave (2x2 tiles): 4 fragment loads -> 4 WMMAs per k-step.
// ---------------------------------------------------------------------------
__device__ __forceinline__ void qkv_store(float* __restrict__ Q,
                                          float* __restrict__ Kt,
                                          float* __restrict__ V,
                                          int o, int s, float val) {
  const int sec = o >> 9, oo = o & 511;                  // 0:Q 1:K 2:V
  const int h = oo >> 6, d = oo & 63;
  if (sec == 0)      Q [(h * PIX + s) * DHEAD + d]  = val;
  else if (sec == 1) Kt[(h * DHEAD + d) * LKV + s]  = val;
  else               V [(h * LKV + s) * DHEAD + d]  = val;
}

__global__ __launch_bounds__(128) void k_qkv_gemm(
    const float* __restrict__ w_qkv, const float* __restrict__ xn,
    float* __restrict__ Q, float* __restrict__ Kt, float* __restrict__ V) {
  const int lane = threadIdx.x & 31, wave = threadIdx.x >> 5;
  const int m = lane & 15, g = lane >> 4;
  const int o0 = (blockIdx.y * 4 + wave) * 32;           // 48 row blocks
  const int s0 = blockIdx.x * 32;                        // 128 col blocks
  v8f acc00 = {}, acc01 = {}, acc10 = {}, acc11 = {};
  const float* a0p = w_qkv + (o0 + m) * C_IN + 2 * g;
  const float* a1p = a0p + 16 * C_IN;
  const float* bp  = xn + (2 * g) * PIX + s0 + m;
  for (int k = 0; k < C_IN; k += 4) {
    v2f a0, a1, b0, b1;
    a0.x = a0p[0];  a0.y = a0p[1];
    a1.x = a1p[0];  a1.y = a1p[1];
    b0.x = bp[0];   b0.y = bp[PIX];
    b1.x = bp[16];  b1.y = bp[PIX + 16];
    acc00 = wmma_f32_k4(a0, b0, acc00);
    acc01 = wmma_f32_k4(a0, b1, acc01);
    acc10 = wmma_f32_k4(a1, b0, acc10);
    acc11 = wmma_f32_k4(a1, b1, acc11);
    a0p += 4; a1p += 4; bp += 4 * PIX;
  }
  for (int r = 0; r < 8; ++r) {
    const int o = o0 + r + 8 * g;
    qkv_store(Q, Kt, V, o,      s0 + m,      acc00[r]);
    qkv_store(Q, Kt, V, o,      s0 + 16 + m, acc01[r]);
    qkv_store(Q, Kt, V, o + 16, s0 + m,      acc10[r]);
    qkv_store(Q, Kt, V, o + 16, s0 + 16 + m, acc11[r]);
  }
}

// ---------------------------------------------------------------------------
// 4) Flash attention.  Block = 4 waves, one 16-query tile each (same head);
//    K/V key-tiles staged in LDS (async global->LDS), P re-laid-out through
//    per-wave LDS (C/D layout -> A layout).
// ---------------------------------------------------------------------------
__global__ __launch_bounds__(128) void k_attn(
    const float* __restrict__ Q, const float* __restrict__ Kt,
    const float* __restrict__ V, float* __restrict__ AO) {
  __shared__ float lds_k[DHEAD * 16];      // [d][key]   4 KB
  __shared__ float lds_v[16 * DHEAD];      // [key][d]   4 KB
  __shared__ float lds_p[4][16 * 16];      // per-wave P 4 KB
  const int lane = threadIdx.x & 31, wave = threadIdx.x >> 5;
  const int m = lane & 15, g = lane >> 4;
  const int h = blockIdx.y;
  const int q0 = (blockIdx.x * 4 + wave) * 16;
  const float scale = 0.125f;              // 64^-0.5

  v2f qa[16];                              // Q tile resident in A-frag form
  for (int kk = 0; kk < 16; ++kk) {
    const float* qp = Q + (h * PIX + q0 + m) * DHEAD + kk * 4 + 2 * g;
    qa[kk].x = qp[0] * scale; qa[kk].y = qp[1] * scale;
  }

  v8f O0 = {}, O1 = {}, O2 = {}, O3 = {};
  float mrow[8], lrow[8];
  for (int r = 0; r < 8; ++r) { mrow[r] = -3.0e38f; lrow[r] = 0.f; }
  float* lp = lds_p[wave];

  for (int jt = 0; jt < KTILES; ++jt) {
    const int kb = jt * 16;
    __syncthreads();
    // prefetch next tile (one cacheline per thread)
    if (jt + 1 < KTILES) {
      const int kb2 = kb + 16;
      if (threadIdx.x < 64)
        __builtin_prefetch(Kt + (h * DHEAD + threadIdx.x) * LKV + kb2, 0, 1);
      else
        __builtin_prefetch(V + (h * LKV + kb2) * DHEAD + (threadIdx.x - 64) * 16, 0, 1);
    }
#if HAVE_ASYNC_LDS
    for (int idx = threadIdx.x; idx < DHEAD * 16; idx += 128) {   // K tile
      const int d = idx >> 4, j = idx & 15;
      lds_async_f32(&lds_k[idx], Kt + (h * DHEAD + d) * LKV + kb + j);
    }
    for (int idx = threadIdx.x; idx < 16 * DHEAD; idx += 128)     // V tile
      lds_async_f32(&lds_v[idx], V + (h * LKV + kb) * DHEAD + idx);
    __builtin_amdgcn_s_wait_asynccnt(0);
#else
    for (int idx = threadIdx.x; idx < DHEAD * 16; idx += 128) {   // K tile
      const int d = idx >> 4, j = idx & 15;
      lds_k[idx] = Kt[(h * DHEAD + d) * LKV + kb + j];
    }
    for (int idx = threadIdx.x; idx < 16 * DHEAD; idx += 128)     // V tile
      lds_v[idx] = V[(h * LKV + kb) * DHEAD + idx];
#endif
    __syncthreads();

    // sim tile = (Q*scale) @ K^T : 16 WMMAs over d=64
    v8f s = {};
    for (int kk = 0; kk < 16; ++kk) {
      v2f b;
      b.x = lds_k[(kk * 4 + 2 * g) * 16 + m];
      b.y = lds_k[(kk * 4 + 2 * g + 1) * 16 + m];
      s = wmma_f32_k4(qa[kk], b, s);
    }
    if (kb + m >= 4100)                      // mask padded key columns
      for (int r = 0; r < 8; ++r) s[r] = -1.0e30f;

    // online softmax (16-lane butterfly reductions)
    float mloc[8];
    for (int r = 0; r < 8; ++r) mloc[r] = s[r];
    for (int off = 8; off >= 1; off >>= 1)
      for (int r = 0; r < 8; ++r)
        mloc[r] = fmaxf(mloc[r], __shfl_xor(mloc[r], off, 32));
    float alpha[8], psum[8];
    for (int r = 0; r < 8; ++r) {
      const float mn = fmaxf(mrow[r], mloc[r]);
      alpha[r] = __expf(mrow[r] - mn);
      mrow[r] = mn;
      const float p = __expf(s[r] - mn);
      s[r] = p; psum[r] = p;
    }
    for (int off = 8; off >= 1; off >>= 1)
      for (int r = 0; r < 8; ++r) psum[r] += __shfl_xor(psum[r], off, 32);
    for (int r = 0; r < 8; ++r) {
      lrow[r] = lrow[r] * alpha[r] + psum[r];
      O0[r] *= alpha[r]; O1[r] *= alpha[r]; O2[r] *= alpha[r]; O3[r] *= alpha[r];
      lp[(r + 8 * g) * 16 + m] = s[r];       // C/D layout -> row-major in LDS
    }
    __syncthreads();

    // O += P @ V_tile : 4 k-steps x 4 d-tiles = 16 WMMAs
    for (int kk = 0; kk < 4; ++kk) {
      v2f a;
      a.x = lp[m * 16 + kk * 4 + 2 * g];
      a.y = lp[m * 16 + kk * 4 + 2 * g + 1];
      const float* vp0 = &lds_v[(kk * 4 + 2 * g) * DHEAD + m];
      const float* vp1 = vp0 + DHEAD;
      v2f b;
      b.x = vp0[0];  b.y = vp1[0];  O0 = wmma_f32_k4(a, b, O0);
      b.x = vp0[16]; b.y = vp1[16]; O1 = wmma_f32_k4(a, b, O1);
      b.x = vp0[32]; b.y = vp1[32]; O2 = wmma_f32_k4(a, b, O2);
      b.x = vp0[48]; b.y = vp1[48]; O3 = wmma_f32_k4(a, b, O3);
    }
  }

  for (int r = 0; r < 8; ++r) {              // normalize + scatter
    const float inv = 1.0f / lrow[r];
    const int q = q0 + r + 8 * g;
    AO[(h * DHEAD +  0 + m) * PIX + q] = O0[r] * inv;
    AO[(h * DHEAD + 16 + m) * PIX + q] = O1[r] * inv;
    AO[(h * DHEAD + 32 + m) * PIX + q] = O2[r] * inv;
    AO[(h * DHEAD + 48 + m) * PIX + q] = O3[r] * inv;
  }
}

// ---------------------------------------------------------------------------
// 5) Output projection: out[256][4096] = w_out[256][512] @ AO[512][4096]
//    Same 2x2 register blocking.
// ---------------------------------------------------------------------------
__global__ __launch_bounds__(128) void k_out_gemm(
    const float* __restrict__ w_out, const float* __restrict__ AO,
    float* __restrict__ out) {
  const int lane = threadIdx.x & 31, wave = threadIdx.x >> 5;
  const int m = lane & 15, g = lane >> 4;
  const int o0 = (blockIdx.y * 4 + wave) * 32;           // 8 row blocks
  const int s0 = blockIdx.x * 32;                        // 128 col blocks
  v8f acc00 = {}, acc01 = {}, acc10 = {}, acc11 = {};
  const float* a0p = w_out + (o0 + m) * HIDDEN + 2 * g;
  const float* a1p = a0p + 16 * HIDDEN;
  const float* bp  = AO + (2 * g) * PIX + s0 + m;
  for (int k = 0; k < HIDDEN; k += 4) {
    v2f a0, a1, b0, b1;
    a0.x = a0p[0];  a0.y = a0p[1];
    a1.x = a1p[0];  a1.y = a1p[1];
    b0.x = bp[0];   b0.y = bp[PIX];
    b1.x = bp[16];  b1.y = bp[PIX + 16];
    acc00 = wmma_f32_k4(a0, b0, acc00);
    acc01 = wmma_f32_k4(a0, b1, acc01);
    acc10 = wmma_f32_k4(a1, b0, acc10);
    acc11 = wmma_f32_k4(a1, b1, acc11);
    a0p += 4; a1p += 4; bp += 4 * PIX;
  }
  for (int r = 0; r < 8; ++r) {
    const int o = o0 + r + 8 * g;
    out[ o       * PIX + s0 + m     ] = acc00[r];
    out[ o       * PIX + s0 + 16 + m] = acc01[r];
    out[(o + 16) * PIX + s0 + m     ] = acc10[r];
    out[(o + 16) * PIX + s0 + 16 + m] = acc11[r];
  }
}

// ---------------------------------------------------------------------------
extern "C" void kernel_launch(void* const* d_in, const int* in_sizes, int n_in,
                              void* d_out, int out_size, void* d_ws,
                              size_t ws_size, hipStream_t stream) {
  (void)in_sizes; (void)n_in; (void)out_size; (void)ws_size;
  const float* x      = (const float*)d_in[0];
  const float* gamma  = (const float*)d_in[1];
  const float* mem_kv = (const float*)d_in[2];
  const float* w_qkv  = (const float*)d_in[3];
  const float* w_out  = (const float*)d_in[4];
  float* out = (float*)d_out;

  // Workspace layout (floats): ~37.8 MB total
  float* ws = (float*)d_ws;
  float* xn = ws;                              // 256*4096
  float* Q  = xn + C_IN * PIX;                 // 8*4096*64
  float* Kt = Q  + HEADS * PIX * DHEAD;        // 8*64*4112
  float* V  = Kt + HEADS * DHEAD * LKV;        // 8*4112*64
  float* AO = V  + HEADS * LKV * DHEAD;        // 512*4096

  k_rmsnorm <<<PIX / 256, 256, 0, stream>>>(x, gamma, xn);
  k_initkv  <<<HEADS, 256, 0, stream>>>(mem_kv, Kt, V);
  k_qkv_gemm<<<dim3(PIX / 32, (QKV_ROWS / 32) / 4), 128, 0, stream>>>(w_qkv, xn, Q, Kt, V);
  k_attn    <<<dim3(PIX / 64, HEADS), 128, 0, stream>>>(Q, Kt, V, AO);
  k_out_gemm<<<dim3(PIX / 32, (C_IN / 32) / 4), 128, 0, stream>>>(w_out, AO, out);
}